// BA_88622355186379
// MI455X (gfx1250) — compile-verified
//
#include <hip/hip_runtime.h>

// Only check on the device pass: AMDGCN builtins are never declared for the
// host (x86) compilation, which is what broke the previous round.
#if defined(__HIP_DEVICE_COMPILE__)
#if !__has_builtin(__builtin_amdgcn_wmma_f32_16x16x4_f32)
#error "gfx1250 f32 WMMA builtin not available on this toolchain (device pass)"
#endif
#endif

typedef __attribute__((ext_vector_type(2))) float v2f;
typedef __attribute__((ext_vector_type(8))) float v8f;

#define D_IN 256
#define D_OUT 64
#define LDSW_STRIDE 260   // 260 mod 64 = 4 -> conflict-free B-frag reads

// ---------------------------------------------------------------------------
// GEMM: pre[n,64] = feat[n,256] @ W.T[256,64] + bias, via V_WMMA_F32_16X16X4_F32
// 8 waves / block; each wave computes a 16x64 output tile.
// ---------------------------------------------------------------------------
__global__ __launch_bounds__(256) void gcn_gemm_wmma(
    const float* __restrict__ feat, const float* __restrict__ W,
    const float* __restrict__ bias, float* __restrict__ pre, int n_nodes)
{
    extern __shared__ float lds_w[];  // D_OUT rows x LDSW_STRIDE floats

    // Cooperative stage of W (64x256 fp32, 64 KB) into LDS (padded rows).
    for (int idx = threadIdx.x; idx < D_OUT * (D_IN / 4); idx += 256) {
        int r  = idx / (D_IN / 4);
        int c4 = (idx % (D_IN / 4)) * 4;
        const float4 w4 = ((const float4*)W)[r * (D_IN / 4) + (c4 >> 2)];
        float* dst = &lds_w[r * LDSW_STRIDE + c4];
        dst[0] = w4.x; dst[1] = w4.y; dst[2] = w4.z; dst[3] = w4.w;
    }
    __syncthreads();

    const int tile     = blockIdx.x * 8 + (threadIdx.x >> 5); // wave-uniform
    const int numTiles = (n_nodes + 15) >> 4;
    if (tile >= numTiles) return;                             // wave-uniform exit

    const int lane = threadIdx.x & 31;
    const int m    = lane & 15;            // A: row-in-tile; B: col-in-ntile
    const int kh   = (lane >> 4) << 1;     // K offset within group-of-4: 0 or 2
    const int row0 = tile << 4;

    int arow_idx = row0 + m;
    if (arow_idx >= n_nodes) arow_idx = n_nodes - 1;          // safe tail clamp
    const float* arow = feat + (size_t)arow_idx * D_IN + kh;

    const float* b0p = &lds_w[(0 * 16 + m) * LDSW_STRIDE + kh];
    const float* b1p = &lds_w[(1 * 16 + m) * LDSW_STRIDE + kh];
    const float* b2p = &lds_w[(2 * 16 + m) * LDSW_STRIDE + kh];
    const float* b3p = &lds_w[(3 * 16 + m) * LDSW_STRIDE + kh];

    v8f acc0 = {}, acc1 = {}, acc2 = {}, acc3 = {};

#pragma unroll 8
    for (int kk = 0; kk < D_IN; kk += 4) {
        v2f a  = *(const v2f*)(arow + kk);
        v2f b0 = *(const v2f*)(b0p + kk);
        v2f b1 = *(const v2f*)(b1p + kk);
        v2f b2 = *(const v2f*)(b2p + kk);
        v2f b3 = *(const v2f*)(b3p + kk);
        acc0 = __builtin_amdgcn_wmma_f32_16x16x4_f32(false, a, false, b0, (short)0, acc0, false, false);
        acc1 = __builtin_amdgcn_wmma_f32_16x16x4_f32(false, a, false, b1, (short)0, acc1, false, false);
        acc2 = __builtin_amdgcn_wmma_f32_16x16x4_f32(false, a, false, b2, (short)0, acc2, false, false);
        acc3 = __builtin_amdgcn_wmma_f32_16x16x4_f32(false, a, false, b3, (short)0, acc3, false, false);
    }

    // C/D layout: VGPR v, lane L -> M = v + 8*(L>>4), N = (L&15) + 16*ntile
    const float bb0 = bias[m], bb1 = bias[16 + m], bb2 = bias[32 + m], bb3 = bias[48 + m];
    const int mbase = (lane >> 4) * 8;
#pragma unroll
    for (int v = 0; v < 8; ++v) {
        int r = row0 + mbase + v;
        if (r < n_nodes) {
            float* o = pre + (size_t)r * D_OUT;
            o[m]      = acc0[v] + bb0;
            o[16 + m] = acc1[v] + bb1;
            o[32 + m] = acc2[v] + bb2;
            o[48 + m] = acc3[v] + bb3;
        }
    }
}

// ---------------------------------------------------------------------------
// Zero the two accumulator surfaces (d_out-as-s and sq) each launch.
// ---------------------------------------------------------------------------
__global__ __launch_bounds__(256) void gcn_zero2(float* __restrict__ a,
                                                 float* __restrict__ b, int n)
{
    int i = blockIdx.x * 256 + threadIdx.x;
    if (i < n) { a[i] = 0.0f; b[i] = 0.0f; }
}

// ---------------------------------------------------------------------------
// Fused double-SpMM scatter: one wave32 per edge; lane handles dims d, d+32.
// s  += v * p      ;  sq += v * p*p     (hardware global_atomic_add_f32)
// ---------------------------------------------------------------------------
__global__ __launch_bounds__(256) void gcn_spmm_scatter(
    const int* __restrict__ rows, const int* __restrict__ cols,
    const float* __restrict__ vals, const float* __restrict__ pre,
    float* __restrict__ s, float* __restrict__ sq, int n_edges)
{
    long gid  = (long)blockIdx.x * 256 + threadIdx.x;
    int  e    = (int)(gid >> 5);       // wave-uniform
    int  lane = (int)(gid & 31);
    if (e >= n_edges) return;

    const int   r = rows[e];
    const int   c = cols[e];
    const float v = vals[e];

    const float* p = pre + (size_t)c * D_OUT;
    float* srow = s  + (size_t)r * D_OUT;
    float* qrow = sq + (size_t)r * D_OUT;

    float p0 = p[lane];
    float p1 = p[lane + 32];
    unsafeAtomicAdd(srow + lane,      v * p0);
    unsafeAtomicAdd(srow + lane + 32, v * p1);
    unsafeAtomicAdd(qrow + lane,      v * p0 * p0);
    unsafeAtomicAdd(qrow + lane + 32, v * p1 * p1);
}

// ---------------------------------------------------------------------------
// out = diag[row] * 0.5 * (s*s - sq)   (s lives in d_out, updated in place)
// ---------------------------------------------------------------------------
__global__ __launch_bounds__(256) void gcn_finalize(
    float* __restrict__ s_out, const float* __restrict__ sq,
    const float* __restrict__ diag, int total)
{
    int i = blockIdx.x * 256 + threadIdx.x;
    if (i < total) {
        float sv = s_out[i];
        s_out[i] = diag[i >> 6] * 0.5f * (sv * sv - sq[i]);
    }
}

extern "C" void kernel_launch(void* const* d_in, const int* in_sizes, int n_in,
                              void* d_out, int out_size, void* d_ws, size_t ws_size,
                              hipStream_t stream)
{
    const float* feat     = (const float*)d_in[0];  // [N, 256]
    const float* W        = (const float*)d_in[1];  // [64, 256]
    const float* bias     = (const float*)d_in[2];  // [64]
    const float* adj_vals = (const float*)d_in[3];  // [E]
    const float* diag     = (const float*)d_in[4];  // [N]
    const int*   adj_rows = (const int*)d_in[5];    // [E]
    const int*   adj_cols = (const int*)d_in[6];    // [E]

    const int n_nodes = in_sizes[4];
    const int n_edges = in_sizes[3];

    float* pre = (float*)d_ws;                        // [N, 64] pre_sup
    float* sq  = pre + (size_t)n_nodes * D_OUT;       // [N, 64] A @ (XW)^2
    float* s   = (float*)d_out;                       // [N, 64] A @ XW, then output

    const int total = n_nodes * D_OUT;

    gcn_zero2<<<(total + 255) / 256, 256, 0, stream>>>(s, sq, total);

    const int tiles  = (n_nodes + 15) / 16;
    const int blocks = (tiles + 7) / 8;
    gcn_gemm_wmma<<<blocks, 256, D_OUT * LDSW_STRIDE * sizeof(float), stream>>>(
        feat, W, bias, pre, n_nodes);

    const long sthreads = (long)n_edges * 32;
    gcn_spmm_scatter<<<(unsigned)((sthreads + 255) / 256), 256, 0, stream>>>(
        adj_rows, adj_cols, adj_vals, pre, s, sq, n_edges);

    gcn_finalize<<<(total + 255) / 256, 256, 0, stream>>>(s, sq, diag, total);
}